// Embedding_3805341024363
// MI455X (gfx1250) — compile-verified
//
#include <hip/hip_runtime.h>

// Embedding gather: out[t, :] = w[x[t], :]
//   x: [8, 2048] int32   (16384 tokens)
//   w: [50257, 1024] f32
// out: [8, 2048, 1024] f32
//
// Memory-bound: ~128 MiB total traffic -> ~5.5us at 23.3 TB/s HBM.
// 256 threads cover one 1024-f32 row as float4 (b128) chunks; each block
// handles 4 rows unrolled (4 outstanding global_load_b128 per lane before
// any store). Indices are block-uniform -> scalar loads (s_load). Output
// uses non-temporal b128 stores so the 64 MiB write stream does not evict
// the gathered w rows from the 192 MB L2 (duplicate tokens then hit L2).
// Hot kernel is branch-free; a tail kernel covers n_tokens % 4 != 0
// (never launched for the reference shape).

typedef __attribute__((ext_vector_type(4))) float v4f;

#define EMB_DIM       1024
#define QDIM          (EMB_DIM / 4)   // 256 float4 per row
#define ROWS_PER_BLK  4
#define BLOCK_THREADS 256

__global__ __launch_bounds__(BLOCK_THREADS) void embed_gather_main(
    const int* __restrict__ x,
    const float* __restrict__ w,
    float* __restrict__ out) {
  const unsigned row0 = blockIdx.x * ROWS_PER_BLK;  // block-uniform
  const unsigned t = threadIdx.x;                   // 0..255 -> float4 column

  const v4f* __restrict__ wq = reinterpret_cast<const v4f*>(w);
  v4f* __restrict__ oq = reinterpret_cast<v4f*>(out);

  // 4 consecutive uniform index loads -> scalar path (s_load_b128).
  unsigned idx[ROWS_PER_BLK];
#pragma unroll
  for (int r = 0; r < ROWS_PER_BLK; ++r) {
    idx[r] = (unsigned)x[row0 + r];
  }

  // 4 independent b128 gathers issue back-to-back (per-wave MLP).
  v4f v[ROWS_PER_BLK];
#pragma unroll
  for (int r = 0; r < ROWS_PER_BLK; ++r) {
    v[r] = wq[(unsigned long long)idx[r] * QDIM + t];
  }

  // Streaming non-temporal b128 stores: output is never re-read.
#pragma unroll
  for (int r = 0; r < ROWS_PER_BLK; ++r) {
    __builtin_nontemporal_store(v[r], &oq[(unsigned long long)(row0 + r) * QDIM + t]);
  }
}

// Tail: one row per block, only launched when n_tokens % ROWS_PER_BLK != 0.
__global__ __launch_bounds__(BLOCK_THREADS) void embed_gather_tail(
    const int* __restrict__ x,
    const float* __restrict__ w,
    float* __restrict__ out,
    unsigned row_base) {
  const unsigned row = row_base + blockIdx.x;
  const unsigned t = threadIdx.x;

  const v4f* __restrict__ wq = reinterpret_cast<const v4f*>(w);
  v4f* __restrict__ oq = reinterpret_cast<v4f*>(out);

  const unsigned idx = (unsigned)x[row];
  v4f v = wq[(unsigned long long)idx * QDIM + t];
  __builtin_nontemporal_store(v, &oq[(unsigned long long)row * QDIM + t]);
}

extern "C" void kernel_launch(void* const* d_in, const int* in_sizes, int n_in,
                              void* d_out, int out_size, void* d_ws, size_t ws_size,
                              hipStream_t stream) {
  const int* x = (const int*)d_in[0];     // [8, 2048] int32
  const float* w = (const float*)d_in[1]; // [50257, 1024] f32
  float* out = (float*)d_out;             // [8, 2048, 1024] f32

  const int n_tokens = in_sizes[0];       // 16384
  const int n_main = n_tokens / ROWS_PER_BLK;
  const int n_tail = n_tokens - n_main * ROWS_PER_BLK;

  if (n_main > 0) {
    embed_gather_main<<<n_main, BLOCK_THREADS, 0, stream>>>(x, w, out);
  }
  if (n_tail > 0) {
    embed_gather_tail<<<n_tail, BLOCK_THREADS, 0, stream>>>(
        x, w, out, (unsigned)(n_main * ROWS_PER_BLK));
  }
}